// GridUnpooling_34265249088352
// MI455X (gfx1250) — compile-verified
//
#include <hip/hip_runtime.h>
#include <hip/hip_bf16.h>

typedef __attribute__((ext_vector_type(2))) float v2f;
typedef __attribute__((ext_vector_type(8))) float v8f;

#define COUT   128
#define KCHUNK 128
#define PITCH  (KCHUNK + 4)   // dword pitch: stride mod 64 banks = 4
#define EPSV   1e-5f

#if __has_builtin(__builtin_amdgcn_sched_group_barrier)
// masks: 0x8 = MFMA/WMMA, 0x20 = VMEM read, 0x100 = DS read
#define SCHED_GROUP(mask, size, id) __builtin_amdgcn_sched_group_barrier(mask, size, id)
#else
#define SCHED_GROUP(mask, size, id)
#endif

// out[m][n] = relu((sum_k A[m][k]*W[n][k] + b[n] - mean[n]) * g[n]/sqrt(v[n]+eps) + beta[n])
//             (+ fcoarse[cluster[m]][n] when GATHER)
// A: [M][K] row-major, W: [128][K] row-major, out: [M][128].  K % KCHUNK == 0.
//
// LDS holds W swizzled into per-WMMA-step quads:
//   quad(2s)   = (k_{8s},   k_{8s+1}, k_{8s+4}, k_{8s+5})
//   quad(2s+1) = (k_{8s+2}, k_{8s+3}, k_{8s+6}, k_{8s+7})
// so a step-q B fragment is float2 at row*PITCH + 4q + 2*half  -> banks {4l+2h, 4l+2h+1},
// conflict-free. The matching A fragment is one float4 at A[row][8s + 4*half]:
// af.xy feeds step 2s, af.zw feeds step 2s+1 (both lane-halves line up with the quads).
template <bool GATHER>
__global__ __launch_bounds__(256) void gridunpool_gemm_bn_relu(
    const float* __restrict__ A, int M, int K,
    const float* __restrict__ W,
    const float* __restrict__ bnb, const float* __restrict__ bng,
    const float* __restrict__ bnbeta, const float* __restrict__ bnm,
    const float* __restrict__ bnv,
    float* __restrict__ out,
    const int* __restrict__ cluster,       // used when GATHER
    const float* __restrict__ fcoarse)     // used when GATHER, [*][128]
{
    extern __shared__ float sW[];          // [128][PITCH] floats

    const int tid  = threadIdx.x;
    const int lane = tid & 31;
    const int wave = tid >> 5;
    const int half = lane >> 4;
    const int lo   = lane & 15;
    const int rowBase = blockIdx.x * 128 + wave * 16;
    const int arow    = rowBase + lo;
    // Clamp instead of predicating: keeps EXEC untouched in the hot loop.
    const float* __restrict__ Arow = A + (size_t)(arow < M ? arow : (M - 1)) * K;

    v8f acc[8] = {};                       // 8 tiles of 16x16 f32 accumulators

    for (int kc = 0; kc < K; kc += KCHUNK) {
        __syncthreads();                   // protect previous chunk's readers
        // Stage + swizzle W[:, kc:kc+KCHUNK] into LDS: 8 source floats -> 2 permuted quads.
        for (int i = tid; i < COUT * (KCHUNK / 8); i += 256) {
            const int n = i >> 4;                  // KCHUNK/8 == 16 groups per row
            const int j = (i & 15) << 3;           // float offset of 8-wide group
            const float4 s0 = *(const float4*)&W[(size_t)n * K + kc + j];      // k j..j+3
            const float4 s1 = *(const float4*)&W[(size_t)n * K + kc + j + 4];  // k j+4..j+7
            float4 q0, q1;
            q0.x = s0.x; q0.y = s0.y; q0.z = s1.x; q0.w = s1.y;  // (j, j+1, j+4, j+5)
            q1.x = s0.z; q1.y = s0.w; q1.z = s1.z; q1.w = s1.w;  // (j+2, j+3, j+6, j+7)
            *(float4*)&sW[n * PITCH + j]     = q0;               // 16B aligned (PITCH%4==0)
            *(float4*)&sW[n * PITCH + j + 4] = q1;
        }
        __syncthreads();

        for (int s = 0; s < KCHUNK / 8; ++s) {
            const float4 af = *(const float4*)&Arow[kc + 8 * s + 4 * half];  // global_load_b128
            v2f a0; a0.x = af.x; a0.y = af.y;      // step 2s
            v2f a1; a1.x = af.z; a1.y = af.w;      // step 2s+1

            float2 w0[8], w1[8];                   // batched conflict-free LDS pair-loads
            #pragma unroll
            for (int t = 0; t < 8; ++t) {
                const float* p = &sW[(t * 16 + lo) * PITCH + 8 * s + 2 * half];
                w0[t] = *(const float2*)p;         // step 2s   fragment
                w1[t] = *(const float2*)(p + 4);   // step 2s+1 fragment
            }

            #pragma unroll
            for (int t = 0; t < 8; ++t) {
                v2f b0; b0.x = w0[t].x; b0.y = w0[t].y;
                acc[t] = __builtin_amdgcn_wmma_f32_16x16x4_f32(
                    false, a0, false, b0, (short)0, acc[t], false, false);
            }
            #pragma unroll
            for (int t = 0; t < 8; ++t) {
                v2f b1; b1.x = w1[t].x; b1.y = w1[t].y;
                acc[t] = __builtin_amdgcn_wmma_f32_16x16x4_f32(
                    false, a1, false, b1, (short)0, acc[t], false, false);
            }

            // Pin issue order: A load first (longest latency), then ALL LDS reads,
            // then the 16 WMMAs -> first WMMA waits at dscnt<=7 instead of 0 per tile.
            SCHED_GROUP(0x020, 1, 0);    // 1x VMEM read  (A b128)
            SCHED_GROUP(0x100, 16, 0);   // all DS reads  (8x 2addr_b64 or 16x b64)
            SCHED_GROUP(0x008, 16, 0);   // 16x WMMA
        }
    }

    // Gather indices: row depends only on (r, half) -> 16 lanes broadcast-load the same int.
    int crow[8];
    if (GATHER) {
        #pragma unroll
        for (int r = 0; r < 8; ++r) {
            const int rr = rowBase + r + 8 * half;
            crow[r] = (rr < M) ? cluster[rr] : 0;
        }
    }

    // Epilogue: folded BN + ReLU (+ gather add). C/D layout: VGPR r holds rows r / r+8.
    #pragma unroll
    for (int t = 0; t < 8; ++t) {
        const int col = t * 16 + lo;
        const float sc  = bng[col] / sqrtf(bnv[col] + EPSV);
        const float ofs = (bnb[col] - bnm[col]) * sc + bnbeta[col];
        #pragma unroll
        for (int r = 0; r < 8; ++r) {
            const int rr = rowBase + r + 8 * half;
            if (rr < M) {
                float val = fmaxf(acc[t][r] * sc + ofs, 0.0f);
                if (GATHER) val += fcoarse[(size_t)crow[r] * COUT + col];
                out[(size_t)rr * COUT + col] = val;
            }
        }
    }
}

// Pass-through: skip_xyz (float4 vectorized) + skip_offset bits into the tail slot.
__global__ void gridunpool_copy_xyz_offset(
    const float* __restrict__ xyz, const int* __restrict__ skip_off,
    float* __restrict__ out, int n, long long off_idx)
{
    const int i  = blockIdx.x * blockDim.x + threadIdx.x;
    const int n4 = n >> 2;
    if (i < n4) ((float4*)out)[i] = ((const float4*)xyz)[i];
    if (i == 0) {
        for (int j = n4 * 4; j < n; ++j) out[j] = xyz[j];   // tail (n%4)
        ((int*)out)[off_idx] = skip_off[0];                 // bit-copy int32 offset
    }
}

extern "C" void kernel_launch(void* const* d_in, const int* in_sizes, int n_in,
                              void* d_out, int out_size, void* d_ws, size_t ws_size,
                              hipStream_t stream) {
    // setup_inputs() order:
    // 0 xyz  1 features  2 offset  3 skip_xyz  4 skip_features  5 skip_offset  6 cluster
    // 7 W1 8 b1 9 g1 10 be1 11 m1 12 v1   13 W2 14 b2 15 g2 16 be2 17 m2 18 v2
    const float* features      = (const float*)d_in[1];
    const float* skip_xyz      = (const float*)d_in[3];
    const float* skip_features = (const float*)d_in[4];
    const int*   skip_offset   = (const int*)d_in[5];
    const int*   cluster       = (const int*)d_in[6];
    const float* W1 = (const float*)d_in[7];
    const float* b1 = (const float*)d_in[8],  *g1 = (const float*)d_in[9];
    const float* be1= (const float*)d_in[10], *m1 = (const float*)d_in[11];
    const float* v1 = (const float*)d_in[12];
    const float* W2 = (const float*)d_in[13];
    const float* b2 = (const float*)d_in[14], *g2 = (const float*)d_in[15];
    const float* be2= (const float*)d_in[16], *m2 = (const float*)d_in[17];
    const float* v2 = (const float*)d_in[18];

    const int n_dec  = in_sizes[0] / 3;          // 125000
    const int n_skip = in_sizes[3] / 3;          // 500000
    const int c_dec  = in_sizes[1] / n_dec;      // 256
    const int c_enc  = in_sizes[4] / n_skip;     // 128

    float* out     = (float*)d_out;
    float* fcoarse = (float*)d_ws;               // [n_dec][128] = 64 MB scratch

    const size_t smem = (size_t)COUT * PITCH * sizeof(float);   // 67584 B (<= 320KB/WGP)

    // fc1: fcoarse = relu(BN(features @ W1^T))   [125000 x 128]
    gridunpool_gemm_bn_relu<false><<<(n_dec + 127) / 128, 256, smem, stream>>>(
        features, n_dec, c_dec, W1, b1, g1, be1, m1, v1, fcoarse, nullptr, nullptr);

    // fc2 + gather: out_f = fcoarse[cluster] + relu(BN(skip_features @ W2^T))
    float* f_out = out + (size_t)n_skip * 3;
    gridunpool_gemm_bn_relu<true><<<(n_skip + 127) / 128, 256, smem, stream>>>(
        skip_features, n_skip, c_enc, W2, b2, g2, be2, m2, v2, f_out, cluster, fcoarse);

    // skip_xyz pass-through + skip_offset bit-copy into last output slot
    const int n_xyz = n_skip * 3;
    const long long off_idx = (long long)n_xyz + (long long)n_skip * COUT;
    const int blocks = ((n_xyz >> 2) + 255) / 256;
    gridunpool_copy_xyz_offset<<<blocks, 256, 0, stream>>>(
        skip_xyz, skip_offset, out, n_xyz, off_idx);
}